// CenterLoss_5222680232363
// MI455X (gfx1250) — compile-verified
//
#include <hip/hip_runtime.h>
#include <hip/hip_bf16.h>

typedef __attribute__((ext_vector_type(2))) float v2f;
typedef __attribute__((ext_vector_type(8))) float v8f;

#define CL_BATCH 2048
#define CL_NUM_CLASSES 100000
#define CL_FEAT 128

// One wave (32 lanes) per 16-row tile. Computes the 16x16 Gram tile
// X_tile * gathered_centers^T via v_wmma_f32_16x16x4_f32 (K=128 in 32 steps),
// takes the diagonal, forms ||x||^2 + ||c||^2 - 2*dot per row, clips, and
// writes the wave's partial sum (16 rows) to partial[tile].
__global__ __launch_bounds__(32)
void centerloss_partial_kernel(const float* __restrict__ x,
                               const int* __restrict__ labels,
                               const float* __restrict__ centers,
                               float* __restrict__ partial)
{
    const int lane = threadIdx.x;       // 0..31
    const int tile = blockIdx.x;        // 0..127
    const int m    = lane & 15;         // row within tile (also N column)
    const int hi   = lane >> 4;         // lane-half selects K sub-pairs
    const int row  = tile * 16 + m;
    const int lab  = labels[row];

    const float* __restrict__ xr = x       + (size_t)row * CL_FEAT;
    const float* __restrict__ cr = centers + (size_t)lab * CL_FEAT;

    v8f acc = {};                 // 16x16 f32 C/D accumulator (8 VGPRs)
    float xsq = 0.0f, csq = 0.0f; // per-lane partial squared norms (half row each)

    // A 16x4 f32 layout: V0 = {lanes0-15:K0, lanes16-31:K2}, V1 = {K1, K3}.
    // B 4x16 f32 uses the same K split with N = lane&15. So every lane loads
    // the float2 at offset 4k + 2*hi from both its x-row and its center-row.
    #pragma unroll 4
    for (int k = 0; k < CL_FEAT / 4; ++k) {
        const int off = 4 * k + 2 * hi;
        v2f a, b;
        a[0] = xr[off]; a[1] = xr[off + 1];
        b[0] = cr[off]; b[1] = cr[off + 1];
        xsq += a[0] * a[0] + a[1] * a[1];
        csq += b[0] * b[0] + b[1] * b[1];
        // (neg_a, A, neg_b, B, c_mod, C, reuse_a, reuse_b)
        acc = __builtin_amdgcn_wmma_f32_16x16x4_f32(
            false, a, false, b, (short)0, acc, false, false);
    }

    // Lane L and L+16 each covered half of row (L&15): fold the halves.
    xsq += __shfl_xor(xsq, 16, 32);
    csq += __shfl_xor(csq, 16, 32);

    // Diagonal D[n][n]: n<8 -> lane n, VGPR n; n>=8 -> lane n+16, VGPR n-8.
    float diag = 0.0f;
    const int sel = lane & 7;
    #pragma unroll
    for (int i = 0; i < 8; ++i)
        if (sel == i) diag = acc[i];

    float v = 0.0f;
    if ((lane < 8) || (lane >= 24)) {
        float d = xsq + csq - 2.0f * diag;   // matches reference formulation
        d = fminf(fmaxf(d, 1e-12f), 1e12f);  // clip after masking
        v = d;
    }

    // Wave32 butterfly sum (inactive lanes contribute 0).
    #pragma unroll
    for (int s = 16; s >= 1; s >>= 1)
        v += __shfl_xor(v, s, 32);

    if (lane == 0)
        partial[tile] = v;
}

// Single-wave finalize: reduce 128 partials, add the clamp constant from the
// B*(C-1) masked zeros, divide by B.
__global__ __launch_bounds__(32)
void centerloss_final_kernel(const float* __restrict__ partial,
                             float* __restrict__ out)
{
    const int lane = threadIdx.x;
    float s = 0.0f;
    #pragma unroll
    for (int i = 0; i < 128; i += 32)
        s += partial[lane + i];
    #pragma unroll
    for (int sh = 16; sh >= 1; sh >>= 1)
        s += __shfl_xor(s, sh, 32);
    if (lane == 0)
        out[0] = s / (float)CL_BATCH + (float)(CL_NUM_CLASSES - 1) * 1e-12f;
}

extern "C" void kernel_launch(void* const* d_in, const int* in_sizes, int n_in,
                              void* d_out, int out_size, void* d_ws, size_t ws_size,
                              hipStream_t stream) {
    (void)in_sizes; (void)n_in; (void)out_size; (void)ws_size;
    const float* x       = (const float*)d_in[0];   // (2048, 128)
    const int*   labels  = (const int*)d_in[1];     // (2048,)
    const float* centers = (const float*)d_in[2];   // (100000, 128)
    float* out     = (float*)d_out;                 // scalar loss
    float* partial = (float*)d_ws;                  // 128 floats of scratch

    centerloss_partial_kernel<<<CL_BATCH / 16, 32, 0, stream>>>(x, labels, centers, partial);
    centerloss_final_kernel<<<1, 32, 0, stream>>>(partial, out);
}